// GraphAttentionTransformerMD17_71554155151890
// MI455X (gfx1250) — compile-verified
//
#include <hip/hip_runtime.h>
#include <math.h>

// ---------------------------------------------------------------------------
// Per-edge RBF featurization (exp-normal smearing * cosine cutoff, + Gaussian
// expansion).  1.02 GB of output stores -> ~45 us HBM floor at 23.3 TB/s.
// Design notes (MI455X / gfx1250, wave32):
//  * one wave per edge, 8 edges per wave (amortize wave setup)
//  * lane owns 4 contiguous RBFs -> single NT b128 store (512 B / wave / instr)
//  * raw v_exp_f32 / v_cos_f32 / v_sqrt_f32 via amdgcn builtins, with all
//    scale constants pre-folded into log2/(2*pi) space (1 instr / transcendental)
//  * edge indices staged to LDS with gfx1250 async-to-LDS (ASYNCcnt) path
// ---------------------------------------------------------------------------

#define WAVES_PER_BLOCK 8
#define ITERS           8
#define EDGES_PER_BLOCK (WAVES_PER_BLOCK * ITERS)   // 64
#define BLOCK_THREADS   256

typedef float v4f __attribute__((ext_vector_type(4)));   // native vector for NT store

constexpr double CUTOFF_D = 6.0;
constexpr int    NUM_RBF  = 128;
constexpr int    GAUSS_NB = 32;
constexpr int    FEAT     = NUM_RBF + GAUSS_NB;                 // 160
constexpr double LOG2E_D  = 1.4426950408889634074;
constexpr double START_D  = 0.0024787521766663585;              // exp(-6.0)
constexpr double STEP_D   = (1.0 - START_D) / (NUM_RBF - 1);    // linspace step
constexpr double BETA_D   = 1.0 / (((2.0 / NUM_RBF) * (1.0 - START_D)) *
                                   ((2.0 / NUM_RBF) * (1.0 - START_D)));
constexpr double ALPHA_D  = 5.0 / CUTOFF_D;
constexpr double GSTEP_D  = 10.0 / (GAUSS_NB - 1);              // gauss center step
constexpr double GW_D     = 0.5 * 10.0 / GAUSS_NB;              // 0.15625
constexpr double GINV_D   = 1.0 / (2.0 * GW_D * GW_D);          // 20.48

// log2-space constants: exp(x) == v_exp_f32(x * log2(e))
constexpr double KRBF_D   = -BETA_D  * LOG2E_D;   // exp(-beta*df^2)  -> exp2(KRBF*df^2)
constexpr double KGAU_D   = -GINV_D  * LOG2E_D;   // exp(-ginv*df^2)  -> exp2(KGAU*df^2)
constexpr double KALP_D   = -ALPHA_D * LOG2E_D;   // exp(-alpha*d)    -> exp2(KALP*d)
// v_cos_f32 input is in revolutions: cos(d*pi/6) == v_cos(d * (pi/6)/(2*pi)) = v_cos(d/12)
constexpr double KCOS_D   = 1.0 / 12.0;

#if defined(__has_builtin)
# if __has_builtin(__builtin_amdgcn_global_load_async_to_lds_b32) && \
     __has_builtin(__builtin_amdgcn_s_wait_asynccnt)
#  define USE_ASYNC_LDS 1
# else
#  define USE_ASYNC_LDS 0
# endif
#else
# define USE_ASYNC_LDS 0
#endif

typedef __attribute__((address_space(1))) int global_i32;
typedef __attribute__((address_space(3))) int lds_i32;

__global__ __launch_bounds__(BLOCK_THREADS)
void rbf_edge_features(const float* __restrict__ pos,
                       const int*   __restrict__ edge_index,  // [2][E]
                       float*       __restrict__ out,         // [E][160]
                       int E)
{
    __shared__ int s_idx[2 * EDGES_PER_BLOCK];   // [0..63]=src, [64..127]=dst

    const int lane = threadIdx.x & 31;
    const int wave = threadIdx.x >> 5;
    const int e0   = blockIdx.x * EDGES_PER_BLOCK;

    // ---- stage this block's 128 edge indices into LDS (async path) --------
    const int t = threadIdx.x;
    if (t < 2 * EDGES_PER_BLOCK) {
        const int which = t >> 6;                      // 0 = src row, 1 = dst row
        const int i     = t & (EDGES_PER_BLOCK - 1);
        int e = e0 + i;
        if (e >= E) e = E - 1;                         // clamp (tail safety)
        const int* gp = edge_index + (size_t)which * (size_t)E + (size_t)e;
#if USE_ASYNC_LDS
        __builtin_amdgcn_global_load_async_to_lds_b32(
            (global_i32*)gp, (lds_i32*)&s_idx[t], 0, 0);
#else
        s_idx[t] = *gp;
#endif
    }
#if USE_ASYNC_LDS
    __builtin_amdgcn_s_wait_asynccnt(0);
#endif
    __syncthreads();

    // ---- per-lane loop invariants -----------------------------------------
    // RBF k = 4*lane + m, m in 0..3 (one float4 per lane)
    const float mean0 = (float)START_D + (float)(4.0 * STEP_D) * (float)lane;
    const float mean1 = mean0 + (float)STEP_D;
    const float mean2 = mean0 + (float)(2.0 * STEP_D);
    const float mean3 = mean0 + (float)(3.0 * STEP_D);
    const float gcen  = (float)GSTEP_D * (float)lane;   // gauss center, k = lane
    const float KR    = (float)KRBF_D;
    const float KG    = (float)KGAU_D;
    const float KA    = (float)KALP_D;

    // ---- 8 edges per wave --------------------------------------------------
    for (int it = 0; it < ITERS; ++it) {
        const int ii = it * WAVES_PER_BLOCK + wave;
        const int e  = e0 + ii;
        if (e >= E) break;                              // wave-uniform branch

        // indices are wave-uniform -> force into SGPRs so pos loads scalarize
        const int s = __builtin_amdgcn_readfirstlane(s_idx[ii]);
        const int d = __builtin_amdgcn_readfirstlane(s_idx[EDGES_PER_BLOCK + ii]);

        const float dx = pos[3 * (size_t)s + 0] - pos[3 * (size_t)d + 0];
        const float dy = pos[3 * (size_t)s + 1] - pos[3 * (size_t)d + 1];
        const float dz = pos[3 * (size_t)s + 2] - pos[3 * (size_t)d + 2];
        const float dist = __builtin_amdgcn_sqrtf(
            fmaf(dx, dx, fmaf(dy, dy, fmaf(dz, dz, 1e-12f))));

        // cosine cutoff: 0.5*(cos(d*pi/6)+1) gated at d<6
        float cc = fmaf(__builtin_amdgcn_cosf(dist * (float)KCOS_D), 0.5f, 0.5f);
        cc = (dist < (float)CUTOFF_D) ? cc : 0.0f;
        // exp-normal argument exp(-alpha*d)
        const float am = __builtin_amdgcn_exp2f(KA * dist);

        // 4 exp-normal RBFs per lane -> one b128 NT store (512 B per wave)
        const float df0 = am - mean0;
        const float df1 = am - mean1;
        const float df2 = am - mean2;
        const float df3 = am - mean3;
        v4f v;
        v.x = cc * __builtin_amdgcn_exp2f((KR * df0) * df0);
        v.y = cc * __builtin_amdgcn_exp2f((KR * df1) * df1);
        v.z = cc * __builtin_amdgcn_exp2f((KR * df2) * df2);
        v.w = cc * __builtin_amdgcn_exp2f((KR * df3) * df3);

        float* __restrict__ row = out + (size_t)e * FEAT;
        __builtin_nontemporal_store(v, (v4f*)(row + 4 * lane));

        // 1 Gaussian feature per lane
        const float dg = dist - gcen;
        __builtin_nontemporal_store(__builtin_amdgcn_exp2f((KG * dg) * dg),
                                    row + NUM_RBF + lane);
    }
}

extern "C" void kernel_launch(void* const* d_in, const int* in_sizes, int n_in,
                              void* d_out, int out_size, void* d_ws, size_t ws_size,
                              hipStream_t stream)
{
    (void)n_in; (void)out_size; (void)d_ws; (void)ws_size;
    const float* pos = (const float*)d_in[0];
    const int*   ei  = (const int*)d_in[1];     // edge_index, [2][E], int32
    float*       out = (float*)d_out;

    const int E      = in_sizes[1] / 2;
    const int blocks = (E + EDGES_PER_BLOCK - 1) / EDGES_PER_BLOCK;

    rbf_edge_features<<<blocks, BLOCK_THREADS, 0, stream>>>(pos, ei, out, E);
}